// ImprovedGraphSAGE_25512105738319
// MI455X (gfx1250) — compile-verified
//
#include <hip/hip_runtime.h>

// GraphSAGE 2-layer forward for MI455X (gfx1250, wave32).
//   layer: out = relu( (segmean x)[i] @ Wl^T + b + x[i] @ Wr^T )
// Aggregation: L2-resident gather + guaranteed hardware global_atomic_add_f32
// (inline asm, non-returning) -- the real roofline limiter (~614M atomics,
// all hitting the 192MB L2 since every operand table fits).
// GEMMs: fp32 WMMA (v_wmma_f32_16x16x4_f32) fed from LDS-staged operands
// (ds_load_b64 fragments, software-pipelined against s_wait_dscnt).

typedef __attribute__((ext_vector_type(2))) float v2f;
typedef __attribute__((ext_vector_type(8))) float v8f;

#define IN_CH 128
#define HID   256

#define KC   32          // K-chunk staged in LDS
#define KCP  36          // padded row stride (16B-aligned, bank-conflict-free)
#define MT   32          // rows per block tile

__device__ __forceinline__ void atom_add_f32(float* p, float v) {
  // Non-returning hardware atomic: single global_atomic_add_f32, no CAS loop.
  // Tracked on STOREcnt; kernel-exit implicit s_wait_idle covers completion.
  unsigned long long a = (unsigned long long)(uintptr_t)p;
  asm volatile("global_atomic_add_f32 %0, %1, off"
               :
               : "v"(a), "v"(v)
               : "memory");
}

__global__ void zero_f32_kernel(float* __restrict__ p, long long n) {
  long long i = (long long)blockIdx.x * blockDim.x + threadIdx.x;
  if (i < n) p[i] = 0.0f;
}

// One thread handles 4 channels of one edge; a wave covers one (layer-1) or
// half of one (layer-2) edge row -> coalesced gather + coalesced atomic
// scatter, all L2-resident.
__global__ void sage_scatter_kernel(const float* __restrict__ x,
                                    const int* __restrict__ src,
                                    const int* __restrict__ dst,
                                    float* __restrict__ s,
                                    float* __restrict__ cnt,
                                    int cshift,          // log2(C/4): 5 or 6
                                    long long nedges, int C) {
  long long tid = (long long)blockIdx.x * blockDim.x + threadIdx.x;
  long long total = nedges << cshift;
  if (tid >= total) return;
  int cg = (int)(tid & ((1 << cshift) - 1));
  long long e = tid >> cshift;
  int si = src[e];
  int di = dst[e];
  const float4 xv =
      *reinterpret_cast<const float4*>(x + (size_t)si * C + (size_t)cg * 4);
  float* p = s + (size_t)di * C + (size_t)cg * 4;
  atom_add_f32(p + 0, xv.x);
  atom_add_f32(p + 1, xv.y);
  atom_add_f32(p + 2, xv.z);
  atom_add_f32(p + 3, xv.w);
  if (cnt != nullptr && cg == 0) atom_add_f32(cnt + di, 1.0f);
}

// Block = 256 threads = 8 waves: 2 row-groups x 4 col-groups.
// Block tile: 32 rows x 256 cols. Wave tile: 16 x 64 = four 16x16 C tiles.
// K stepped in LDS chunks of 32; both matmul terms (mean path and root path)
// fused into the same accumulators. ~81KB LDS -> ~3 blocks resident per WGP.
template <int K>
__global__ __launch_bounds__(256) void sage_gemm_kernel(
    const float* __restrict__ s, const float* __restrict__ cnt,
    const float* __restrict__ xr, const float* __restrict__ Wl,
    const float* __restrict__ Wr, const float* __restrict__ bias,
    float* __restrict__ out, int nrows) {
  __shared__ float lds_wl[HID * KCP];   // weight chunk, n-major
  __shared__ float lds_wr[HID * KCP];
  __shared__ float lds_s[MT * KCP];     // mean operand (pre-scaled by 1/deg)
  __shared__ float lds_x[MT * KCP];

  const int tid = threadIdx.x;
  const int lane = tid & 31;
  const int wv = tid >> 5;
  const int wv_m = wv >> 2;             // 0..1  row group (16 rows)
  const int wv_n = wv & 3;              // 0..3  col group (64 cols)
  const int row_base = blockIdx.x * MT;

  // A-fragment addressing (16x4 fp32): lanes 0-15 hold row=lane, K=k0..k0+1;
  // lanes 16-31 hold row=lane-16, K=k0+2..k0+3 -> contiguous b64 loads.
  const int lrow = lane & 15;
  const int koff = (lane >> 4) << 1;    // 0 or 2
  const int n0 = wv_n * 64 + lrow;      // global col this lane owns per tile

  v8f acc0 = {}, acc1 = {}, acc2 = {}, acc3 = {};

  for (int k0 = 0; k0 < K; k0 += KC) {
    __syncthreads();
    // ---- stage Wl/Wr chunk: 256 rows x 32 k, 8 float4 per thread each ----
#pragma unroll
    for (int i = 0; i < 8; ++i) {
      const int idx = tid + i * 256;    // 0..2047
      const int n = idx >> 3;           // 0..255
      const int c4 = (idx & 7) << 2;    // 0,4,..,28
      *reinterpret_cast<float4*>(&lds_wl[n * KCP + c4]) =
          *reinterpret_cast<const float4*>(Wl + (size_t)n * K + k0 + c4);
      *reinterpret_cast<float4*>(&lds_wr[n * KCP + c4]) =
          *reinterpret_cast<const float4*>(Wr + (size_t)n * K + k0 + c4);
    }
    // ---- stage A chunk: 32 rows x 32 k, 1 float4 per thread each ----
    {
      const int r = tid >> 3;           // 0..31
      const int c4 = (tid & 7) << 2;    // 0..28
      int m = row_base + r;
      if (m >= nrows) m = nrows - 1;    // clamp (stores are guarded later)
      const float invc = 1.0f / fmaxf(cnt[m], 1.0f);
      float4 sv =
          *reinterpret_cast<const float4*>(s + (size_t)m * K + k0 + c4);
      sv.x *= invc; sv.y *= invc; sv.z *= invc; sv.w *= invc;
      *reinterpret_cast<float4*>(&lds_s[r * KCP + c4]) = sv;
      *reinterpret_cast<float4*>(&lds_x[r * KCP + c4]) =
          *reinterpret_cast<const float4*>(xr + (size_t)m * K + k0 + c4);
    }
    __syncthreads();

    // ---- compute: 8 k-steps of 4, all operands from LDS ----
    const float* as_base = &lds_s[(wv_m * 16 + lrow) * KCP + koff];
    const float* ax_base = &lds_x[(wv_m * 16 + lrow) * KCP + koff];
#pragma unroll
    for (int ks = 0; ks < KC / 4; ++ks) {
      const int kk = ks * 4;
      const v2f a_s = *reinterpret_cast<const v2f*>(as_base + kk);
      const v2f a_x = *reinterpret_cast<const v2f*>(ax_base + kk);
#define SAGE_TILE(ACC, T)                                                    \
      {                                                                      \
        const int nl = wv_n * 64 + (T) * 16 + lrow;                          \
        const v2f b_l =                                                      \
            *reinterpret_cast<const v2f*>(&lds_wl[nl * KCP + kk + koff]);    \
        const v2f b_r =                                                      \
            *reinterpret_cast<const v2f*>(&lds_wr[nl * KCP + kk + koff]);    \
        ACC = __builtin_amdgcn_wmma_f32_16x16x4_f32(                         \
            false, a_s, false, b_l, (short)0, ACC, false, false);            \
        ACC = __builtin_amdgcn_wmma_f32_16x16x4_f32(                         \
            false, a_x, false, b_r, (short)0, ACC, false, false);            \
      }
      SAGE_TILE(acc0, 0)
      SAGE_TILE(acc1, 1)
      SAGE_TILE(acc2, 2)
      SAGE_TILE(acc3, 3)
#undef SAGE_TILE
    }
  }

  // C/D layout: VGPR i -> row (i or i+8 for upper lane half), col = lane&15.
  const int mtop = row_base + wv_m * 16 + ((lane >> 4) << 3);
#define SAGE_EPI(ACC, T)                                              \
  {                                                                   \
    const int n = n0 + (T) * 16;                                      \
    const float bv = bias[n];                                         \
    _Pragma("unroll") for (int i = 0; i < 8; ++i) {                   \
      const int r = mtop + i;                                         \
      if (r < nrows) {                                                \
        float v = ACC[i] + bv;                                        \
        out[(size_t)r * HID + n] = v > 0.0f ? v : 0.0f;               \
      }                                                               \
    }                                                                 \
  }
  SAGE_EPI(acc0, 0)
  SAGE_EPI(acc1, 1)
  SAGE_EPI(acc2, 2)
  SAGE_EPI(acc3, 3)
#undef SAGE_EPI
}

extern "C" void kernel_launch(void* const* d_in, const int* in_sizes, int n_in,
                              void* d_out, int out_size, void* d_ws,
                              size_t ws_size, hipStream_t stream) {
  const float* x   = (const float*)d_in[0];
  const int*  eidx = (const int*)d_in[1];  // harness convention: integer -> int*
  const float* Wl1 = (const float*)d_in[2];
  const float* b1  = (const float*)d_in[3];
  const float* Wr1 = (const float*)d_in[4];
  const float* Wl2 = (const float*)d_in[5];
  const float* b2  = (const float*)d_in[6];
  const float* Wr2 = (const float*)d_in[7];

  const long long n_nodes = in_sizes[0] / IN_CH;
  const long long n_edges = in_sizes[1] / 2;
  const int* src = eidx;
  const int* dst = eidx + n_edges;

  // Workspace: s1 [N,128] | cnt [N] | s2 [N,256]   (~77 MB, L2-resident)
  float* ws  = (float*)d_ws;
  float* s1  = ws;
  float* cnt = s1 + (size_t)n_nodes * IN_CH;
  float* s2  = cnt + (size_t)n_nodes;
  float* out = (float*)d_out;  // h1 lives here between layers; layer 2 in-place

  const long long zn = n_nodes * IN_CH + n_nodes + (long long)n_nodes * HID;
  zero_f32_kernel<<<(unsigned)((zn + 255) / 256), 256, 0, stream>>>(ws, zn);

  const int gemm_blocks = (int)((n_nodes + MT - 1) / MT);

  {  // layer 1 aggregate: 32 channel-groups/edge; also compute degrees
    const long long tot = n_edges << 5;
    sage_scatter_kernel<<<(unsigned)((tot + 255) / 256), 256, 0, stream>>>(
        x, src, dst, s1, cnt, 5, n_edges, IN_CH);
  }
  sage_gemm_kernel<IN_CH><<<gemm_blocks, 256, 0, stream>>>(
      s1, cnt, x, Wl1, Wr1, b1, out, (int)n_nodes);

  {  // layer 2 aggregate over h1 (in d_out): 64 channel-groups/edge
    const long long tot = n_edges << 6;
    sage_scatter_kernel<<<(unsigned)((tot + 255) / 256), 256, 0, stream>>>(
        out, src, dst, s2, nullptr, 6, n_edges, HID);
  }
  sage_gemm_kernel<HID><<<gemm_blocks, 256, 0, stream>>>(
      s2, cnt, out, Wl2, Wr2, b2, out, (int)n_nodes);
}